// GraphAttentionLayer_84207128805823
// MI455X (gfx1250) — compile-verified
//
#include <hip/hip_runtime.h>

// GAT layer: B=32, N=1024, IN_C=64, OUT_C=128
//   h  = X @ W^T   (stored transposed: h_t[b][d][n], d-major)
//   e  = leakyrelu(s1[i]+s2[j]) * G ; 0 -> -1e16 ; fused online softmax
//   out= att @ h + bias
// Matmuls on V_WMMA_F32_16X16X4_F32. Attention kernel: 4 waves/block share a
// double-buffered LDS h-tile filled with global_load_async_to_lds_b128 (ASYNCcnt).

#define B_    32
#define N_    1024
#define INC   64
#define OUTC  128
#define ALPHA 0.2f
#define NEGL  (-1e16f)

#define LPITCH 20                 // LDS dword pitch per d-row (bank-conflict-free, 8B aligned)

typedef __attribute__((ext_vector_type(2))) float v2f;
typedef __attribute__((ext_vector_type(8))) float v8f;

// ---------------------------------------------------------------------------
// Kernel 1: h_t tile = (X(16x64) @ W^T(64x128))^T via WMMA f32 16x16x4,
// plus s1 = h.a1, s2 = h.a2 via half-wave reductions.
// Fragment layouts (ISA 7.12.2):
//   A 16x4 : lane -> row m=lane&15 ; v0 = A[m][k+2*half], v1 = A[m][k+1+2*half]
//   B 4x16 : v0 = B[k+2*half][n=lane&15], v1 = B[k+1+2*half][n]
//   C/D    : VGPR r -> row r + 8*half, col = lane&15 (per 16-col tile)
// ---------------------------------------------------------------------------
__global__ __launch_bounds__(32)
void gat_proj(const float* __restrict__ X, const float* __restrict__ W,
              const float* __restrict__ a, float* __restrict__ ht,
              float* __restrict__ s1, float* __restrict__ s2)
{
    const int tiles = N_ / 16;
    const int blk   = blockIdx.x;
    const int b     = blk / tiles;
    const int i0    = (blk % tiles) * 16;
    const int lane  = threadIdx.x & 31;
    const int half  = lane >> 4;
    const int lm    = lane & 15;

    const float* Xt = X + ((size_t)b * N_ + i0 + lm) * INC;
    v8f acc[8] = {};

#pragma unroll 4
    for (int kc = 0; kc < INC / 4; ++kc) {
        const int k = kc * 4 + 2 * half;
        v2f af; af.x = Xt[k]; af.y = Xt[k + 1];
        const float* Wc = W + (size_t)lm * INC + k;
#pragma unroll
        for (int dt = 0; dt < 8; ++dt) {
            v2f bf; bf.x = Wc[(size_t)dt * 16 * INC];
            bf.y = Wc[(size_t)dt * 16 * INC + 1];
            acc[dt] = __builtin_amdgcn_wmma_f32_16x16x4_f32(
                false, af, false, bf, (short)0, acc[dt], false, false);
        }
    }

    // store h transposed: h_t[b][d][n]; 8 consecutive n per (dt,half) -> 2x b128
    float* hb = ht + (size_t)b * N_ * OUTC;
#pragma unroll
    for (int dt = 0; dt < 8; ++dt) {
        float* p = hb + (size_t)(dt * 16 + lm) * N_ + i0 + 8 * half;
        float4 lo = make_float4(acc[dt][0], acc[dt][1], acc[dt][2], acc[dt][3]);
        float4 hi = make_float4(acc[dt][4], acc[dt][5], acc[dt][6], acc[dt][7]);
        *reinterpret_cast<float4*>(p)     = lo;
        *reinterpret_cast<float4*>(p + 4) = hi;
    }

    // s1/s2: per-lane partial dot along columns n = dt*16+lm, reduce over halves
    float a1v[8], a2v[8];
#pragma unroll
    for (int dt = 0; dt < 8; ++dt) {
        a1v[dt] = a[dt * 16 + lm];
        a2v[dt] = a[OUTC + dt * 16 + lm];
    }
#pragma unroll
    for (int r = 0; r < 8; ++r) {
        float p1 = 0.f, p2 = 0.f;
#pragma unroll
        for (int dt = 0; dt < 8; ++dt) {
            p1 += acc[dt][r] * a1v[dt];
            p2 += acc[dt][r] * a2v[dt];
        }
#pragma unroll
        for (int off = 1; off < 16; off <<= 1) {
            p1 += __shfl_xor(p1, off, 32);
            p2 += __shfl_xor(p2, off, 32);
        }
        if (lm == 0) {
            s1[(size_t)b * N_ + i0 + r + 8 * half] = p1;
            s2[(size_t)b * N_ + i0 + r + 8 * half] = p2;
        }
    }
}

// ---------------------------------------------------------------------------
// Kernel 2: fused masked-softmax attention. Block = 4 waves x 16 query rows.
// The 16x128 h-tile for each j-chunk is staged once per block into LDS by the
// async DMA path (double buffered) and consumed by all 4 waves as b64 fragments.
// ---------------------------------------------------------------------------
__global__ __launch_bounds__(128)
void gat_attn(const float* __restrict__ G, const float* __restrict__ bvec,
              const float* __restrict__ ht, const float* __restrict__ s1g,
              const float* __restrict__ s2g, float* __restrict__ out)
{
    __shared__ float hbuf[2][OUTC * LPITCH];   // 2 x 10 KB

    const int rowblks = N_ / 64;
    const int blk  = blockIdx.x;
    const int b    = blk / rowblks;
    const int i0   = (blk % rowblks) * 64 + (threadIdx.x >> 5) * 16; // wave's 16 rows
    const int tid  = threadIdx.x;
    const int lane = tid & 31;
    const int half = lane >> 4;
    const int lm   = lane & 15;

    const float* hb = ht + (size_t)b * N_ * OUTC;   // h_t[b]: [128][1024]

    // issue async copy of h_t[:, j0:j0+16] -> hbuf[bi] ([128][LPITCH], b128 chunks)
    auto issue = [&](int bi, int j0) {
#pragma unroll
        for (int it = 0; it < 4; ++it) {
            const int idx = it * 128 + tid;      // 512 x 16B segments
            const int d = idx >> 2, q = idx & 3;
            const float* src = hb + (size_t)d * N_ + j0 + 4 * q;
            const float* dst = &hbuf[bi][d * LPITCH + 4 * q];
            asm volatile("global_load_async_to_lds_b128 %0, %1, off"
                         :: "v"((unsigned)(uintptr_t)dst),
                            "v"((unsigned long long)(uintptr_t)src)
                         : "memory");
        }
    };

    v8f acc[8] = {};
    float rowmax = -INFINITY;
    float rowsum = 0.0f;

    const float  s1m   = s1g[(size_t)b * N_ + i0 + lm];
    const float* Grow  = G + (size_t)(i0 + lm) * N_;
    const float* s2row = s2g + (size_t)b * N_;

    issue(0, 0);                                  // prologue: chunk 0 -> buf 0

    for (int c = 0; c < N_ / 16; ++c) {
        const int cur = c & 1;
        const int j0  = c * 16;

        if (c + 1 < N_ / 16) {
            issue(1 - cur, j0 + 16);              // next chunk -> other buffer
            asm volatile("s_wait_asynccnt 4" ::: "memory");  // current chunk landed
        } else {
            asm volatile("s_wait_asynccnt 0" ::: "memory");
        }
        __syncthreads();                          // publish all waves' segments

        // ---- e tile in A-fragment order ----
        float ev[8];
#pragma unroll
        for (int kc = 0; kc < 4; ++kc) {
            const int j = j0 + kc * 4 + 2 * half;
            const float2 g2  = *reinterpret_cast<const float2*>(Grow + j);
            const float2 s2v = *reinterpret_cast<const float2*>(s2row + j);
            float v0 = s1m + s2v.x; v0 = v0 > 0.f ? v0 : ALPHA * v0; v0 *= g2.x;
            float v1 = s1m + s2v.y; v1 = v1 > 0.f ? v1 : ALPHA * v1; v1 *= g2.y;
            ev[2 * kc]     = (v0 == 0.0f) ? NEGL : v0;
            ev[2 * kc + 1] = (v1 == 0.0f) ? NEGL : v1;
        }

        // ---- online softmax stats (row owned by lane&15, mirrored via xor16) ----
        float tmax = ev[0];
#pragma unroll
        for (int k = 1; k < 8; ++k) tmax = fmaxf(tmax, ev[k]);
        tmax = fmaxf(tmax, __shfl_xor(tmax, 16, 32));
        const float newmax = fmaxf(rowmax, tmax);
        const float scale  = __expf(rowmax - newmax);
        rowmax = newmax;

        float pv[8];
        float tsum = 0.f;
#pragma unroll
        for (int k = 0; k < 8; ++k) { pv[k] = __expf(ev[k] - rowmax); tsum += pv[k]; }
        tsum += __shfl_xor(tsum, 16, 32);
        rowsum = rowsum * scale + tsum;

        if (__any(scale != 1.0f)) {
#pragma unroll
            for (int r = 0; r < 8; ++r) {
                const float sc = __shfl(scale, r + 8 * half, 32);
#pragma unroll
                for (int dt = 0; dt < 8; ++dt) acc[dt][r] *= sc;
            }
        }

        // ---- acc += P(16x16) @ htile ; B-frags are single ds_load_b64 each ----
        const float* lb = &hbuf[cur][0];
#pragma unroll
        for (int kc = 0; kc < 4; ++kc) {
            v2f af; af.x = pv[2 * kc]; af.y = pv[2 * kc + 1];
            const int jo = 4 * kc + 2 * half;     // even -> 8B aligned
#pragma unroll
            for (int dt = 0; dt < 8; ++dt) {
                const float2 hv = *reinterpret_cast<const float2*>(
                    lb + (dt * 16 + lm) * LPITCH + jo);
                v2f bf; bf.x = hv.x; bf.y = hv.y;
                acc[dt] = __builtin_amdgcn_wmma_f32_16x16x4_f32(
                    false, af, false, bf, (short)0, acc[dt], false, false);
            }
        }
        __syncthreads();                          // all waves done with hbuf[cur]
    }

    // ---- epilogue: divide by row sum, add bias, store ----
    float inv[8];
#pragma unroll
    for (int r = 0; r < 8; ++r)
        inv[r] = 1.0f / __shfl(rowsum, r + 8 * half, 32);

    float* Ot = out + ((size_t)b * N_ + i0) * OUTC;
#pragma unroll
    for (int dt = 0; dt < 8; ++dt) {
        const float bias = bvec[dt * 16 + lm];
#pragma unroll
        for (int r = 0; r < 8; ++r)
            Ot[(size_t)(r + 8 * half) * OUTC + dt * 16 + lm] = acc[dt][r] * inv[r] + bias;
    }
}

// ---------------------------------------------------------------------------
extern "C" void kernel_launch(void* const* d_in, const int* in_sizes, int n_in,
                              void* d_out, int out_size, void* d_ws, size_t ws_size,
                              hipStream_t stream)
{
    (void)in_sizes; (void)n_in; (void)out_size; (void)ws_size;
    const float* X  = (const float*)d_in[0];   // (B,N,64)
    const float* G  = (const float*)d_in[1];   // (N,N)
    const float* W  = (const float*)d_in[2];   // (128,64)
    const float* bv = (const float*)d_in[3];   // (128,)
    const float* a  = (const float*)d_in[4];   // (256,1)
    float* out = (float*)d_out;                // (B,N,128)

    float* ht = (float*)d_ws;                      // transposed h: B*128*N f32
    float* s1 = ht + (size_t)B_ * N_ * OUTC;
    float* s2 = s1 + (size_t)B_ * N_;

    gat_proj<<<dim3(B_ * (N_ / 16)), 32, 0, stream>>>(X, W, a, ht, s1, s2);
    gat_attn<<<dim3(B_ * (N_ / 64)), 128, 0, stream>>>(G, bv, ht, s1, s2, out);
}